// NN_17_31585189494807
// MI455X (gfx1250) — compile-verified
//
#include <hip/hip_runtime.h>
#include <stdint.h>

// ---------------------------------------------------------------------------
// FSRS-style recurrent scan on MI455X (gfx1250, wave32).
// One thread per batch column; serial t-loop in-thread.
//
// CDNA5 design:
//  * Per-WAVE private LDS staging via global_load_async_to_lds_b32 (ASYNCcnt):
//    producer wave == consumer wave  =>  ZERO barriers in the 256-step loop;
//    s_wait_asynccnt alone orders LDS visibility.
//  * 4-deep ring buffer, async issue 2 steps ahead (wait <= 6 with 3
//    loads/step: in-order retirement guarantees step-t data landed).
//  * Dense 128 B per async instruction per wave (lane i -> dword i of slice).
//  * Non-temporal stores for the 67 MB write-once output stream.
//  * global_prefetch (via __builtin_prefetch) pulls t+4 toward GL2, extending
//    the pipeline one more stage (HBM -> L2 -> LDS -> regs).
//  * No WMMA on purpose: the MLPs are width-3 dot products inside a serial
//    dependency chain (<1% WMMA tile utilization); VALU FMA + trans ops win.
// ---------------------------------------------------------------------------

#define TSTEPS 256
#define BLK    128                 // 4 wave32s per workgroup, 256 workgroups
#define NWAVES (BLK / 32)
#define S_MIN_ 0.01f
#define S_MAX_ 36500.0f

typedef __attribute__((ext_vector_type(2))) float v2f;  // NT-store friendly

__device__ __forceinline__ float clampf(float x, float lo, float hi) {
    return fminf(fmaxf(x, lo), hi);
}

// mish(h) = h * tanh(softplus(h)) = h * (e^2 + 2e) / (e^2 + 2e + 2), e = e^h.
// Clamp only inside exp: for h > 30 the ratio is 1.0 at fp32 precision.
__device__ __forceinline__ float mishf(float h) {
    float e = __expf(fminf(h, 30.0f));
    float n = fmaf(e, e, 2.0f * e);
    return h * __fdividef(n, n + 2.0f);
}

__device__ __forceinline__ float softplusf_(float x) {
    float e  = __expf(fminf(x, 30.0f));
    float sp = log1pf(e);
    return (x > 20.0f) ? x : sp;
}

__device__ __forceinline__ float sigmoidf_(float x) {
    return __fdividef(1.0f, 1.0f + __expf(-x));
}

__device__ __forceinline__ float mlp3(float a, float b, float c, const float p[6]) {
    float h = fmaf(a, p[0], fmaf(b, p[1], fmaf(c, p[2], p[3])));
    return fmaf(mishf(h), p[4], p[5]);
}

__device__ __forceinline__ float mlp2(float a, float b, const float p[5]) {
    float h = fmaf(a, p[0], fmaf(b, p[1], p[2]));
    return fmaf(mishf(h), p[3], p[4]);
}

// Issue this wave's 384 B slice of step t_ into LDS ring slot q_.
// 96 dwords per wave slice, 3 dense async b32 loads (128 B each).
// GVS addressing: mem = SADDR(64b sgpr pair) + VADDR(32b vgpr offset).
#define ISSUE_WAVE_SLICE(t_, q_)                                               \
    do {                                                                       \
        unsigned gslice_ = ((unsigned)(t_) * (unsigned)B + (unsigned)b0) * 12u \
                           + (unsigned)w * 384u;                               \
        unsigned lslice_ = ldsw + (unsigned)(q_) * (NWAVES * 384u);            \
        _Pragma("unroll")                                                      \
        for (int k_ = 0; k_ < 3; ++k_) {                                       \
            unsigned off_ = (unsigned)(k_ * 32 + li) * 4u;                     \
            unsigned la_  = lslice_ + off_;                                    \
            unsigned ga_  = gslice_ + off_;                                    \
            asm volatile(                                                      \
                "global_load_async_to_lds_b32 %0, %1, %2 offset:0"             \
                :: "v"(la_), "v"(ga_), "s"(gbase)                              \
                : "memory");                                                   \
        }                                                                      \
    } while (0)

__global__ __launch_bounds__(BLK) void fsrs_scan_kernel(
    const float* __restrict__ inputs,   // (T, B, 3): delta_t, rating, lapses
    const float* __restrict__ S0g, const float* __restrict__ D0g,
    const float* __restrict__ sincw,
    const float* __restrict__ rwp_g, const float* __restrict__ ndp_g,
    const float* __restrict__ plp_g, const float* __restrict__ snp_g,
    const float* __restrict__ bsp_g,
    float* __restrict__ outs,           // (T, B, 2)
    float* __restrict__ finals,         // (B, 2)
    int B)
{
    // Ring of 4 step-slices; each wave owns a disjoint 384 B region per slice.
    __shared__ float tile[4][NWAVES][96];   // 6 KB

    const int tid = threadIdx.x;
    const int li  = tid & 31;               // lane in wave
    const int w   = tid >> 5;               // wave id in block
    const int b0  = blockIdx.x * BLK;
    const int b   = b0 + tid;

    // Tiny parameter vectors -> registers (uniform -> s_load + broadcast).
    float S0r[4], D0r[4], rwp[6], ndp[6], snp[6], plp[5], bsp[5];
#pragma unroll
    for (int i = 0; i < 4; ++i) { S0r[i] = S0g[i]; D0r[i] = D0g[i]; }
#pragma unroll
    for (int i = 0; i < 6; ++i) { rwp[i] = rwp_g[i]; ndp[i] = ndp_g[i]; snp[i] = snp_g[i]; }
#pragma unroll
    for (int i = 0; i < 5; ++i) { plp[i] = plp_g[i]; bsp[i] = bsp_g[i]; }
    const float w0e = __expf(sincw[0]);     // exp(sinc_w[0]) hoisted
    const float w1  = sincw[1];
    const float w2  = sincw[2];

    const unsigned long long gbase = (unsigned long long)(uintptr_t)inputs;
    // Byte address of this wave's region in slice 0.
    const unsigned ldsw = (unsigned)(uintptr_t)&tile[0][w][0];

    // Prologue: fill pipeline two steps deep.
    ISSUE_WAVE_SLICE(0, 0);
    ISSUE_WAVE_SLICE(1, 1);

    float s_prev = 1.0f, d_prev = 1.0f;

    for (int t = 0; t < TSTEPS; ++t) {
        // Issue step t+2 into ring slot (t+2)&3. Slot (t+2)&3 was last read at
        // step t-2 by THIS wave (consumed before this point in program order),
        // so per-wave reuse is race-free without any barrier.
        const int nt = (t + 2 < TSTEPS) ? (t + 2) : (TSTEPS - 1);
        ISSUE_WAVE_SLICE(nt, (t + 2) & 3);

        // Stage t+4 toward GL2 (speculative stream prefetch).
        const int pt = (t + 4 < TSTEPS) ? (t + 4) : (TSTEPS - 1);
        __builtin_prefetch(inputs + ((size_t)pt * (size_t)B + (size_t)b) * 3, 0, 1);

        // 3 async loads per step, retired in order per wave:
        // <=6 outstanding  =>  step-t slice has landed in LDS.
        asm volatile("s_wait_asynccnt 0x6" ::: "memory");

        const float* tp = &tile[t & 3][w][li * 3];
        const float dt     = tp[0];
        const float rating = tp[1];
        const float lapses = tp[2];

        float new_s, new_d;
        if (t == 0) {
            // cond = all(state == 1.0): exactly true only at t==0 (state0=ones);
            // for t>=1 it would need every sigmoid output == 1.0f over all B.
            int  ri   = (int)rating;
            bool mask = (ri >= 1) && (ri <= 4);
            int  idx  = min(max(ri - 1, 0), 3);
            float si  = mask ? S0r[idx] : 0.0f;
            float di  = mask ? D0r[idx] : 0.0f;
            new_s = clampf(si, S_MIN_, S_MAX_);
            new_d = di;
        } else {
            // rt = clip(0.9^(dt/s), 1e-4, 0.9999) = clip(exp(ln0.9*dt/s), ..)
            float rt = __expf(-0.10536051565782628f * __fdividef(dt, s_prev));
            rt = clampf(rt, 0.0001f, 0.9999f);

            // rw = clip(exp(-softplus(h)), ..) = clip(1/(1+e^h), ..)
            // log(rw_clipped) = -clamp(softplus(h), -ln0.9999, -ln1e-4)
            float h_rw = mlp3(d_prev, s_prev, rt, rwp);
            float eh   = __expf(fminf(h_rw, 30.0f));
            float rw   = clampf(__fdividef(1.0f, 1.0f + eh), 0.0001f, 0.9999f);
            float sp   = (h_rw > 20.0f) ? h_rw : log1pf(eh);
            sp = clampf(sp, 1.0000500033e-4f, 9.2103403719761836f);

            // sr = clip(ln0.9/log(rw) * dt, S_MIN, S_MAX) = 0.105360../sp * dt
            float sr = clampf(__fdividef(0.10536051565782628f, sp) * dt,
                              S_MIN_, S_MAX_);

            float nd  = sigmoidf_(mlp3(d_prev, rw, rating, ndp));
            float pls = clampf(softplusf_(mlp2(rw, lapses, plp)), S_MIN_, S_MAX_);

            // sinc_t = 1 + e^{w0} * (5(1-nd)+1) * sr^{-w1} * e^{-rw*w2}
            float sinc_t = fmaf(
                w0e * fmaf(5.0f, 1.0f - nd, 1.0f),
                __expf(-w1 * __logf(sr)) * __expf(-rw * w2), 1.0f);
            float sinc_nn = 1.0f + softplusf_(mlp3(nd, sr, rw, snp));
            float best    = 1.0f + softplusf_(mlp2(sinc_t, sinc_nn, bsp));

            float s_rec = (rating > 1.0f) ? sr * best : pls;
            new_s = clampf(s_rec, S_MIN_, S_MAX_);
            new_d = nd;
        }

        // Coalesced 8 B/lane streaming store: outs[t, b, :] = {s, d}.
        // Write-once stream -> non-temporal, keep the 192 MB L2 for inputs.
        v2f v; v.x = new_s; v.y = new_d;
        __builtin_nontemporal_store(
            v, reinterpret_cast<v2f*>(
                   outs + ((size_t)t * (size_t)B + (size_t)b) * 2));

        s_prev = new_s;
        d_prev = new_d;
    }

    v2f v; v.x = s_prev; v.y = d_prev;
    __builtin_nontemporal_store(
        v, reinterpret_cast<v2f*>(finals + (size_t)b * 2));
}

extern "C" void kernel_launch(void* const* d_in, const int* in_sizes, int n_in,
                              void* d_out, int out_size, void* d_ws, size_t ws_size,
                              hipStream_t stream) {
    (void)in_sizes; (void)n_in; (void)d_ws; (void)ws_size;

    const float* inputs = (const float*)d_in[0];
    const float* S0v    = (const float*)d_in[1];
    const float* D0v    = (const float*)d_in[2];
    const float* sinc_w = (const float*)d_in[3];
    const float* rw_p   = (const float*)d_in[4];
    const float* nd_p   = (const float*)d_in[5];
    const float* pls_p  = (const float*)d_in[6];
    const float* sinc_p = (const float*)d_in[7];
    const float* best_p = (const float*)d_in[8];

    // out = outs (T*B*2) ++ final (B*2)  =>  B = out_size / (2*T + 2)
    const int B = out_size / (2 * TSTEPS + 2);

    float* outs   = (float*)d_out;
    float* finals = outs + (size_t)TSTEPS * (size_t)B * 2;

    dim3 grid((unsigned)(B / BLK));
    dim3 block(BLK);
    hipLaunchKernelGGL(fsrs_scan_kernel, grid, block, 0, stream,
                       inputs, S0v, D0v, sinc_w, rw_p, nd_p, pls_p, sinc_p,
                       best_p, outs, finals, B);
}